// SmoothHoleFiller_71708773974041
// MI455X (gfx1250) — compile-verified
//
#include <hip/hip_runtime.h>
#include <math.h>

typedef __attribute__((ext_vector_type(2))) float v2f;
typedef __attribute__((ext_vector_type(8))) float v8f;

#define KH 480
#define KW 640
#define IMGSZ (KH * KW)
#define NIMG 8
#define TX 40                 // tiles per row   (640/16)
#define TY 30                 // tiles per col   (480/16)
#define TPI (TX * TY)
#define XR 24                 // input tile rows  (16 + 6 halo, padded to 24 for K%4)
#define XP 40                 // input tile pitch (32 cols used; pitch 40 -> conflict free)
#define TP 25                 // T1 pitch (24 cols used; odd pitch -> conflict free)
#define NWAVE 4
#define ITERS 50

struct WaveLds {
    float X  [XR][XP];        // filled values  (24 x 32 valid)
    float Xm [XR][XP];        // validity mask
    float T1s[16][TP];        // vertical-conv of values (16 x 24)
    float T1c[16][TP];        // vertical-conv of mask
    float tab[40];            // zero-padded banded-coefficient table: tab[d+15] = g1[d]
};

__global__ __launch_bounds__(NWAVE * 32)
void holefill_iter(const float* __restrict__ src_all,
                   const float* __restrict__ sparse_all,
                   const float* __restrict__ g2,      // 7x7 normalized gaussian (OIHW)
                   float* __restrict__ dst_all)
{
    __shared__ WaveLds lds[NWAVE];
    const int wave = threadIdx.x >> 5;
    const int lane = threadIdx.x & 31;
    WaveLds& Lw = lds[wave];

    const int tile = blockIdx.x * NWAVE + wave;
    const int img  = tile / TPI;
    if (img >= NIMG) return;                     // whole-wave exit (grid is exact anyway)
    const int rem  = tile - img * TPI;
    const int trow = rem / TX;
    const int tcol = rem - trow * TX;
    const int row0 = trow * 16;
    const int col0 = tcol * 16;

    const float* __restrict__ src = src_all    + img * IMGSZ;
    const float* __restrict__ spd = sparse_all + img * IMGSZ;
    float*       __restrict__ dst = dst_all    + img * IMGSZ;

    const int mrow  = lane & 15;   // M (stage1-A) == N (stage2-B) index for this lane
    const int khalf = lane >> 4;   // which K half-pair this lane holds

    // ---- Banded coefficient table in LDS (per wave; no cross-wave sync needed) --
    // g2 = g1n (x) g1n  =>  g1n[j] = g2[3][j] / sqrt(g2[3][3]);  g2 row 3 = g2[21..27].
    // tab[i] = g1n[i-15] for i in [15,21], else 0.  Needed i range: [0,38].
    {
        const float inv = 1.0f / sqrtf(g2[3 * 7 + 3]);
        float tv = 0.0f;
        if (lane >= 15 && lane <= 21) tv = g2[6 + lane] * inv;
        Lw.tab[lane] = tv;
        if (lane < 8) Lw.tab[32 + lane] = 0.0f;
    }

    // Band[m,k] = g1[k-m] (0<=k-m<=6).  Identical lane formula serves both the
    // stage-1 A operand and the stage-2 B operand (32-bit WMMA fragment layouts).
    v2f coef[6];
    #pragma unroll
    for (int s = 0; s < 6; ++s) {
        const int base = 4 * s + 2 * khalf - mrow + 15;   // in [0, 37]
        coef[s][0] = Lw.tab[base + 0];
        coef[s][1] = Lw.tab[base + 1];
    }

    // ---- Load 24x32 input tile, build mask tile ---------------------------------
    // Interior tiles (~89%) skip all boundary clamping; condition is wave-uniform.
    if (trow >= 1 && trow <= TY - 2 && tcol >= 1 && tcol <= TX - 2) {
        const float* p0 = src + (row0 - 3) * KW + (col0 - 3) + lane;
        #pragma unroll
        for (int r = 0; r < XR; ++r) {
            const float v = p0[r * KW];
            Lw.X [r][lane] = v;
            Lw.Xm[r][lane] = (v != 0.0f) ? 1.0f : 0.0f;
        }
    } else {
        const int c   = col0 - 3 + lane;
        const bool ci = (c >= 0) && (c < KW);
        #pragma unroll
        for (int r = 0; r < XR; ++r) {
            const int rr = row0 - 3 + r;
            float v = 0.0f;
            if (ci && rr >= 0 && rr < KH) v = src[rr * KW + c];
            Lw.X [r][lane] = v;
            Lw.Xm[r][lane] = (v != 0.0f) ? 1.0f : 0.0f;
        }
    }

    // ---- Stage 1: vertical banded matmul  T1(16x24) = Band(16x24) x X(24x32) ----
    #pragma unroll
    for (int ch = 0; ch < 2; ++ch) {
        const float (*Xt)[XP] = ch ? Lw.Xm : Lw.X;
        float (*T1)[TP]       = ch ? Lw.T1c : Lw.T1s;
        #pragma unroll
        for (int nt = 0; nt < 2; ++nt) {
            v8f acc = {0.f, 0.f, 0.f, 0.f, 0.f, 0.f, 0.f, 0.f};
            #pragma unroll
            for (int s = 0; s < 6; ++s) {
                v2f b;                                   // B frag: [k, n] -> lane n+16*(k>=2)
                b[0] = Xt[4 * s + 2 * khalf + 0][16 * nt + mrow];
                b[1] = Xt[4 * s + 2 * khalf + 1][16 * nt + mrow];
                acc = __builtin_amdgcn_wmma_f32_16x16x4_f32(
                        false, coef[s], false, b, (short)0, acc, false, false);
            }
            const int cc = 16 * nt + mrow;               // D frag -> T1 LDS (keep cols < 24)
            #pragma unroll
            for (int r = 0; r < 8; ++r) {
                if (cc < 24) T1[r + 8 * khalf][cc] = acc[r];
            }
        }
    }

    // ---- Stage 2: horizontal banded matmul  out(16x16) = T1(16x24) x Band^T -----
    v8f sum = {0.f, 0.f, 0.f, 0.f, 0.f, 0.f, 0.f, 0.f};
    v8f cnt = {0.f, 0.f, 0.f, 0.f, 0.f, 0.f, 0.f, 0.f};
    #pragma unroll
    for (int s = 0; s < 6; ++s) {
        v2f a, c;                                        // A frag: [m, k] -> lane m+16*(k>=2)
        a[0] = Lw.T1s[mrow][4 * s + 2 * khalf + 0];
        a[1] = Lw.T1s[mrow][4 * s + 2 * khalf + 1];
        c[0] = Lw.T1c[mrow][4 * s + 2 * khalf + 0];
        c[1] = Lw.T1c[mrow][4 * s + 2 * khalf + 1];
        sum = __builtin_amdgcn_wmma_f32_16x16x4_f32(
                false, a, false, coef[s], (short)0, sum, false, false);
        cnt = __builtin_amdgcn_wmma_f32_16x16x4_f32(
                false, c, false, coef[s], (short)0, cnt, false, false);
    }

    // ---- Epilogue: safe divide + select against original sparse input -----------
    #pragma unroll
    for (int r = 0; r < 8; ++r) {
        const int m   = r + 8 * khalf;                   // D layout: lanes 16-31 hold M=r+8
        const int row = row0 + m;
        const int col = col0 + mrow;
        const float s0 = spd[row * KW + col];
        const float c0 = cnt[r];
        const float avg = (c0 > 0.0f) ? (sum[r] / c0) : 0.0f;
        dst[row * KW + col] = (s0 != 0.0f) ? s0 : avg;
    }
}

extern "C" void kernel_launch(void* const* d_in, const int* in_sizes, int n_in,
                              void* d_out, int out_size, void* d_ws, size_t ws_size,
                              hipStream_t stream)
{
    const float* sparse = (const float*)d_in[0];   // (8,1,480,640) fp32
    const float* g2     = (const float*)d_in[1];   // (1,1,7,7)     fp32
    float* out = (float*)d_out;
    float* ws  = (float*)d_ws;                     // needs 8 images = 9.83 MB

    const dim3 grid((NIMG * TPI) / NWAVE);         // 2400 blocks, exact cover
    const dim3 block(NWAVE * 32);

    // Ping-pong ws <-> out; parity makes iteration 50 (it=49, odd) write d_out.
    for (int it = 0; it < ITERS; ++it) {
        const float* src = (it == 0) ? sparse : ((it & 1) ? ws : out);
        float*       dst = (it & 1) ? out : ws;
        holefill_iter<<<grid, block, 0, stream>>>(src, sparse, g2, dst);
    }
}